// BandSplitDecoder_83863531421924
// MI455X (gfx1250) — compile-verified
//
#include <hip/hip_runtime.h>
#include <math.h>

#define NBANDS 32
#define HIDDEN 256
#define TT 512
#define BB 8
#define NFREQS 1025
#define MTOT (BB * TT)      /* 4096 rows = (b,t) */
#define TILE_M 128
#define TILE_N 64
#define TILE_K 32

typedef __attribute__((ext_vector_type(2))) float v2f;
typedef __attribute__((ext_vector_type(8))) float v8f;

struct BandArgs {
    const float* W[NBANDS];     // (HIDDEN, od) row-major
    const float* bias[NBANDS];  // (od)
    int s[NBANDS];              // start freq bin
    int w[NBANDS];              // band width; od = 4*w
};

#define A_STRIDE  36    // 32 + pad (floats)
#define BP_STRIDE 130   // one K-pair row: 64 cols * 2 + pad (floats)
#define C_STRIDE  65    // 64 + pad (floats)
// A: 128*36 = 4608 floats, B: 16*130 = 2080 floats -> 6688 floats (26.8 KB)
// C (epilogue reuse): 128*65 = 8320 floats (33.3 KB)
#define SMEM_FLOATS 8320

__global__ __launch_bounds__(256)
void bandsplit_decoder_kernel(const float* __restrict__ bands,
                              float* __restrict__ out,
                              BandArgs args)
{
    __shared__ float smem[SMEM_FLOATS];
    float* As = smem;                       // [128][A_STRIDE]           m x k
    float* Bs = smem + TILE_M * A_STRIDE;   // [16][BP_STRIDE] K-pair interleaved

    const int band  = blockIdx.z;
    const int sfreq = args.s[band];
    const int wband = args.w[band];
    const int od    = 4 * wband;
    const int nbase = blockIdx.x * TILE_N;
    if (nbase >= od) return;                // uniform exit
    const int mbase = blockIdx.y * TILE_M;

    const float* __restrict__ Wp = args.W[band];
    const float* __restrict__ bp = args.bias[band];

    const int tid  = threadIdx.x;
    const int lane = tid & 31;
    const int wave = tid >> 5;   // 0..7 -> rows 16*wave .. 16*wave+15

    // All 128 rows of this tile share one batch index (512 % 128 == 0).
    const int bidx = mbase >> 9;      // / TT
    const int t0   = mbase & 511;     // % TT
    const float* __restrict__ Ag =
        bands + (((size_t)bidx * NBANDS + band) * TT + t0) * HIDDEN;

    v8f acc0 = {};   // cols  0..15
    v8f acc1 = {};   // cols 16..31
    v8f acc2 = {};   // cols 32..47
    v8f acc3 = {};   // cols 48..63

    const int hi   = lane >> 4;       // 0: lanes 0-15, 1: lanes 16-31
    const int l16  = lane & 15;
    const int arow = 16 * wave + l16;
    const float* __restrict__ bbase = Bs + (l16 << 1);

    for (int k0 = 0; k0 < HIDDEN; k0 += TILE_K) {
        // ---- stage A tile (128 x 32) : 1024 float4, 4 per thread ----
        #pragma unroll
        for (int q = tid; q < 1024; q += 256) {
            const int row = q >> 3;          // 8 float4 per row
            const int c4  = (q & 7) << 2;
            const float4 v = *(const float4*)(Ag + (size_t)row * HIDDEN + k0 + c4);
            float* dst = As + row * A_STRIDE + c4;
            dst[0] = v.x; dst[1] = v.y; dst[2] = v.z; dst[3] = v.w;
        }
        // ---- stage B tile (32 x 64), K-pair interleaved: Bs[k/2][col*2 + (k&1)] ----
        #pragma unroll
        for (int q = tid; q < 512; q += 256) {
            const int row = q >> 4;          // k-local, 0..31
            const int c4  = (q & 15) << 2;   // n-local
            const int n   = nbase + c4;
            float4 v;
            if (n < od) {                    // od is a multiple of 4 -> no straddle
                v = *(const float4*)(Wp + (size_t)(k0 + row) * od + n);
            } else {
                v = make_float4(0.f, 0.f, 0.f, 0.f);
            }
            float* dst = Bs + (row >> 1) * BP_STRIDE + (row & 1) + (c4 << 1);
            dst[0] = v.x; dst[2] = v.y; dst[4] = v.z; dst[6] = v.w;
        }
        __syncthreads();

        // ---- 8 K-steps of V_WMMA_F32_16X16X4_F32, 4 N sub-tiles each ----
        #pragma unroll
        for (int kk = 0; kk < TILE_K; kk += 4) {
            // A 16x4 fragment: lanes 0-15 -> K = kk,kk+1 ; lanes 16-31 -> K = kk+2,kk+3
            const int akk = kk + (hi << 1);
            const v2f a = *(const v2f*)(As + arow * A_STRIDE + akk);
            // B fragments: one aligned b64 per column group (K pair contiguous)
            const float* bsrc = bbase + ((kk >> 1) + hi) * BP_STRIDE;
            const v2f b0 = *(const v2f*)(bsrc + 0);
            const v2f b1 = *(const v2f*)(bsrc + 32);
            const v2f b2 = *(const v2f*)(bsrc + 64);
            const v2f b3 = *(const v2f*)(bsrc + 96);

            acc0 = __builtin_amdgcn_wmma_f32_16x16x4_f32(
                       false, a, false, b0, (short)0, acc0, false, false);
            acc1 = __builtin_amdgcn_wmma_f32_16x16x4_f32(
                       false, a, false, b1, (short)0, acc1, false, false);
            acc2 = __builtin_amdgcn_wmma_f32_16x16x4_f32(
                       false, a, false, b2, (short)0, acc2, false, false);
            acc3 = __builtin_amdgcn_wmma_f32_16x16x4_f32(
                       false, a, false, b3, (short)0, acc3, false, false);
        }
        __syncthreads();
    }

    // ---- epilogue: transpose accumulators through LDS so lanes run along t ----
    float* Cs = smem;  // [128][C_STRIDE], overlaps As/Bs (safe after barrier)
    const int crow = 16 * wave + (hi ? 8 : 0);  // C/D layout: vgpr r -> M=r (+8 hi lanes)
    const int ccol = l16;
    #pragma unroll
    for (int r = 0; r < 8; ++r) {
        Cs[(crow + r) * C_STRIDE + ccol]      = acc0[r];
        Cs[(crow + r) * C_STRIDE + ccol + 16] = acc1[r];
        Cs[(crow + r) * C_STRIDE + ccol + 32] = acc2[r];
        Cs[(crow + r) * C_STRIDE + ccol + 48] = acc3[r];
    }
    __syncthreads();

    const int w2 = 2 * wband;
    for (int idx = tid; idx < TILE_M * TILE_N; idx += 256) {
        const int trow  = idx & 127;    // consecutive lanes -> consecutive t
        const int n     = idx >> 7;
        const int nglob = nbase + n;
        if (nglob < od) {
            const int ri  = nglob / w2;          // 0 = real, 1 = imag
            const int rem = nglob - ri * w2;
            const int c   = rem / wband;         // channel
            const int j   = rem - c * wband;     // bin within band
            const int f   = sfreq + j;
            const float val = tanhf(Cs[trow * C_STRIDE + n] + bp[nglob]);
            const int t = t0 + trow;
            const size_t o =
                2u * ((((size_t)bidx * 2 + c) * NFREQS + f) * TT + t) + ri;
            out[o] = val;
        }
    }
}

// Zero-fill frequency bins never written by any band (tanh(0) == 0).
__global__ void bandsplit_fill_kernel(float* __restrict__ out, int fstart)
{
    const int nf    = NFREQS - fstart;
    const int total = BB * 2 * nf * TT;   // complex elements
    const int i = blockIdx.x * blockDim.x + threadIdx.x;
    if (i < total) {
        const int t = i % TT;
        int r = i / TT;
        const int f = fstart + (r % nf); r /= nf;
        const int c = r % 2;
        const int b = r / 2;
        const size_t o = 2u * ((((size_t)b * 2 + c) * NFREQS + f) * TT + t);
        out[o]     = 0.f;
        out[o + 1] = 0.f;
    }
}

extern "C" void kernel_launch(void* const* d_in, const int* in_sizes, int n_in,
                              void* d_out, int out_size, void* d_ws, size_t ws_size,
                              hipStream_t stream)
{
    (void)in_sizes; (void)n_in; (void)d_ws; (void)ws_size;
    const float* bands = (const float*)d_in[0];
    float* out = (float*)d_out;

    // Recompute mel band edges exactly as the reference (deterministic, host-side).
    const double maxf   = 44100.0 / 2.0;
    const double melmax = 2595.0 * log10(1.0 + maxf / 700.0);
    int edges[NBANDS + 1];
    for (int i = 0; i <= NBANDS; ++i) {
        const double m  = melmax * (double)i / (double)NBANDS;
        const double hz = 700.0 * (pow(10.0, m / 2595.0) - 1.0);
        edges[i] = (int)llround(hz / maxf * (double)(NFREQS - 1));
    }

    BandArgs args;
    int maxod = 0, fend = 0, prev_end = 0;
    bool gap = false;
    for (int i = 0; i < NBANDS; ++i) {
        int s = edges[i];     if (s > NFREQS - 1) s = NFREQS - 1;
        int e = edges[i + 1]; if (e > NFREQS)     e = NFREQS;
        if (e <= s) e = s + 1;
        args.s[i] = s;
        args.w[i] = e - s;
        args.W[i]    = (const float*)d_in[1 + 2 * i];
        args.bias[i] = (const float*)d_in[2 + 2 * i];
        const int od = 4 * (e - s);
        if (od > maxod) maxod = od;
        if (s > prev_end) gap = true;
        if (e > prev_end) prev_end = e;
        if (e > fend) fend = e;
    }

    if (gap) {
        // Defensive: with these constants bands tile [0,1024) contiguously,
        // but if a gap ever appears, zero everything first.
        hipMemsetAsync(d_out, 0, (size_t)out_size * sizeof(float), stream);
    }

    const int ntiles = (maxod + TILE_N - 1) / TILE_N;
    dim3 grid(ntiles, MTOT / TILE_M, NBANDS);
    bandsplit_decoder_kernel<<<grid, 256, 0, stream>>>(bands, out, args);

    if (fend < NFREQS) {
        const int nfill = BB * 2 * (NFREQS - fend) * TT;
        bandsplit_fill_kernel<<<(nfill + 255) / 256, 256, 0, stream>>>(out, fend);
    }
}